// GeometricPreserver_26439818674532
// MI455X (gfx1250) — compile-verified
//
#include <hip/hip_runtime.h>
#include <hip/hip_bf16.h>
#include <math.h>

typedef __attribute__((ext_vector_type(2))) float v2f;
typedef __attribute__((ext_vector_type(8))) float v8f;

#define C_CLASSES 100
#define DIMS      128
#define CPAD      112      // 7 * 16 padded class count for WMMA tiling
#define NB1       128      // blocks for segment-sum pass
#define NB3       128      // blocks for spread pass
#define EPSF      1e-8f

// ---------------------------------------------------------------------------
// K1: per-block partial segment sums (deterministic: thread t owns dim t)
// ---------------------------------------------------------------------------
__global__ void k1_seg_sums(const float* __restrict__ emb,
                            const int* __restrict__ labels, int N,
                            float* __restrict__ psums, float* __restrict__ pcnts) {
    __shared__ float acc[C_CLASSES * DIMS];   // 51.2 KB
    __shared__ float cnt[C_CLASSES];
    const int t = threadIdx.x;                // 0..127 (dim owner)
    for (int i = t; i < C_CLASSES * DIMS; i += DIMS) acc[i] = 0.f;
    if (t < C_CLASSES) cnt[t] = 0.f;
    __syncthreads();

    const int chunk = (N + gridDim.x - 1) / gridDim.x;
    const int i0 = blockIdx.x * chunk;
    const int i1 = (i0 + chunk < N) ? (i0 + chunk) : N;

    #pragma unroll 4
    for (int i = i0; i < i1; ++i) {
        __builtin_prefetch(emb + (size_t)(i + 8) * DIMS + t, 0, 0);
        const int lab = labels[i];
        acc[lab * DIMS + t] += emb[(size_t)i * DIMS + t];
        if (t == 0) cnt[lab] += 1.f;
    }
    __syncthreads();

    float* out = psums + (size_t)blockIdx.x * (C_CLASSES * DIMS);
    for (int i = t; i < C_CLASSES * DIMS; i += DIMS) out[i] = acc[i];
    if (t < C_CLASSES) pcnts[blockIdx.x * C_CLASSES + t] = cnt[t];
}

// ---------------------------------------------------------------------------
// K2: reduce partials -> centroids, normalized padded centroids, counts
// grid = CPAD blocks (one per padded class), 128 threads (one per dim)
// ---------------------------------------------------------------------------
__global__ void k2_centroids(const float* __restrict__ psums,
                             const float* __restrict__ pcnts,
                             float* __restrict__ centroids,
                             float* __restrict__ cn,
                             float* __restrict__ counts) {
    const int c = blockIdx.x;
    const int d = threadIdx.x;
    __shared__ float red[DIMS];
    __shared__ float s_count;

    if (c >= C_CLASSES) { cn[c * DIMS + d] = 0.f; return; }   // zero padding rows

    float s = 0.f;
    for (int b = 0; b < NB1; ++b)
        s += psums[(size_t)b * (C_CLASSES * DIMS) + c * DIMS + d];

    if (d == 0) {
        float cv = 0.f;
        for (int b = 0; b < NB1; ++b) cv += pcnts[b * C_CLASSES + c];
        s_count = cv;
        counts[c] = cv;
    }
    __syncthreads();

    const float cv   = s_count;
    const float safe = fmaxf(cv, 1.f);
    const float cent = (cv > 0.f) ? (s / safe) : 0.f;
    centroids[c * DIMS + d] = cent;

    red[d] = cent * cent;
    __syncthreads();
    for (int off = DIMS / 2; off > 0; off >>= 1) {
        if (d < off) red[d] += red[d + off];
        __syncthreads();
    }
    const float nrm = sqrtf(red[0]);
    cn[c * DIMS + d] = cent / fmaxf(nrm, EPSF);
}

// ---------------------------------------------------------------------------
// K3: per-block partial spread sums. One wave32 per sample, float4 per lane.
// ---------------------------------------------------------------------------
__global__ void k3_spread(const float* __restrict__ emb,
                          const int* __restrict__ labels,
                          const float* __restrict__ centroids, int N,
                          float* __restrict__ pspread) {
    __shared__ float sp[8 * C_CLASSES];       // per-wave private bins
    const int t = threadIdx.x;                // 0..255 (8 waves)
    const int w = t >> 5, lane = t & 31;
    for (int i = t; i < 8 * C_CLASSES; i += 256) sp[i] = 0.f;
    __syncthreads();

    const int chunk = (N + gridDim.x - 1) / gridDim.x;
    const int i0 = blockIdx.x * chunk;
    const int i1 = (i0 + chunk < N) ? (i0 + chunk) : N;

    for (int i = i0 + w; i < i1; i += 8) {
        const int lab = labels[i];
        const float4 e  = ((const float4*)(emb + (size_t)i * DIMS))[lane];
        const float4 cc = ((const float4*)(centroids + lab * DIMS))[lane];
        const float dx = e.x - cc.x, dy = e.y - cc.y;
        const float dz = e.z - cc.z, dw = e.w - cc.w;
        float s = dx * dx + dy * dy + dz * dz + dw * dw;
        #pragma unroll
        for (int off = 16; off > 0; off >>= 1) s += __shfl_xor(s, off, 32);
        if (lane == 0) sp[w * C_CLASSES + lab] += sqrtf(s);
    }
    __syncthreads();

    for (int c = t; c < C_CLASSES; c += 256) {
        float s = 0.f;
        #pragma unroll
        for (int ww = 0; ww < 8; ++ww) s += sp[ww * C_CLASSES + c];
        pspread[blockIdx.x * C_CLASSES + c] = s;
    }
}

// ---------------------------------------------------------------------------
// K4: finalize spreads (fixed-order reduction over NB3 partials)
// ---------------------------------------------------------------------------
__global__ void k4_spreads(const float* __restrict__ pspread,
                           const float* __restrict__ counts,
                           float* __restrict__ spreads) {
    const int c = threadIdx.x;
    if (c >= C_CLASSES) return;
    float s = 0.f;
    for (int b = 0; b < NB3; ++b) s += pspread[b * C_CLASSES + c];
    const float cv = counts[c];
    spreads[c] = (cv > 0.f) ? (s / fmaxf(cv, 1.f)) : 0.f;
}

// ---------------------------------------------------------------------------
// K5: Gram matrix sims = Cn * Cn^T via V_WMMA_F32_16X16X4_F32.
// One wave per 16x16 tile; 7x7 tiles over padded 112x128 Cn.
// fp32 A-layout: lane<16 -> K+{0,1}; lane>=16 -> K+{2,3} (ISA 7.12.2).
// ---------------------------------------------------------------------------
__global__ void k5_gram_wmma(const float* __restrict__ cn,
                             float* __restrict__ simsP) {
    const int mt = blockIdx.x / (CPAD / 16);
    const int nt = blockIdx.x % (CPAD / 16);
    const int lane = threadIdx.x;             // 0..31, EXEC all ones
    const int m0 = mt * 16, n0 = nt * 16;
    const int rowA = m0 + (lane & 15);
    const int rowB = n0 + (lane & 15);
    const int koff = (lane < 16) ? 0 : 2;

    v8f acc = {};
    #pragma unroll 4
    for (int k0 = 0; k0 < DIMS; k0 += 4) {
        v2f a, b;
        a.x = cn[rowA * DIMS + k0 + koff];
        a.y = cn[rowA * DIMS + k0 + koff + 1];
        b.x = cn[rowB * DIMS + k0 + koff];
        b.y = cn[rowB * DIMS + k0 + koff + 1];
        acc = __builtin_amdgcn_wmma_f32_16x16x4_f32(
            /*neg_a=*/false, a, /*neg_b=*/false, b,
            /*c_mod=*/(short)0, acc, /*reuse_a=*/false, /*reuse_b=*/false);
    }

    // D layout: VGPR r, lanes 0-15 -> M=r, lanes 16-31 -> M=r+8; N = lane&15
    const int col   = n0 + (lane & 15);
    const int mbase = m0 + ((lane < 16) ? 0 : 8);
    #pragma unroll
    for (int r = 0; r < 8; ++r)
        simsP[(mbase + r) * CPAD + col] = acc[r];
}

// ---------------------------------------------------------------------------
// K6: final scalar loss (single block, two phases with block reductions)
// ---------------------------------------------------------------------------
__device__ __forceinline__ float block_reduce_256(float v, float* red) {
    const int t = threadIdx.x;
    red[t] = v;
    __syncthreads();
    for (int off = 128; off > 0; off >>= 1) {
        if (t < off) red[t] += red[t + off];
        __syncthreads();
    }
    const float r = red[0];
    __syncthreads();
    return r;
}

__global__ void k6_loss(const float* __restrict__ simsP,
                        const float* __restrict__ spreads,
                        const float* __restrict__ ref_d,
                        const float* __restrict__ ref_a,
                        const float* __restrict__ ref_s,
                        float* __restrict__ out) {
    __shared__ float red[256];
    const int t = threadIdx.x;
    const int NCC = C_CLASSES * C_CLASSES;

    // Phase 1: sums for means + angle MSE (needs no normalization)
    float sRefD = 0.f, sCurD = 0.f, sAng = 0.f;
    for (int idx = t; idx < NCC; idx += 256) {
        const int r = idx / C_CLASSES, c = idx % C_CLASSES;
        const float sim = simsP[r * CPAD + c];
        sRefD += ref_d[idx];
        sCurD += (1.f - sim);
        const float da = sim - ref_a[idx];
        sAng += da * da;
    }
    float sRefS = 0.f, sCurS = 0.f;
    for (int c = t; c < C_CLASSES; c += 256) { sRefS += ref_s[c]; sCurS += spreads[c]; }

    const float S_refd = block_reduce_256(sRefD, red);
    const float S_curd = block_reduce_256(sCurD, red);
    const float S_ang  = block_reduce_256(sAng,  red);
    const float S_refs = block_reduce_256(sRefS, red);
    const float S_curs = block_reduce_256(sCurS, red);

    // Phase 2: mean-normalized MSEs
    const float mRefD = S_refd / (float)NCC + EPSF;
    const float mCurD = S_curd / (float)NCC + EPSF;
    float sDist = 0.f;
    for (int idx = t; idx < NCC; idx += 256) {
        const int r = idx / C_CLASSES, c = idx % C_CLASSES;
        const float cd = 1.f - simsP[r * CPAD + c];
        const float dd = cd / mCurD - ref_d[idx] / mRefD;
        sDist += dd * dd;
    }
    const float mRefS = S_refs / (float)C_CLASSES + EPSF;
    const float mCurS = S_curs / (float)C_CLASSES + EPSF;
    float sSpr = 0.f;
    for (int c = t; c < C_CLASSES; c += 256) {
        const float ds = spreads[c] / mCurS - ref_s[c] / mRefS;
        sSpr += ds * ds;
    }
    const float S_dist = block_reduce_256(sDist, red);
    const float S_spr  = block_reduce_256(sSpr,  red);

    if (t == 0) {
        out[0] = 1.0f * (S_dist / (float)NCC)
               + 1.0f * (S_ang  / (float)NCC)
               + 0.5f * (S_spr  / (float)C_CLASSES);
    }
}

// ---------------------------------------------------------------------------
extern "C" void kernel_launch(void* const* d_in, const int* in_sizes, int n_in,
                              void* d_out, int out_size, void* d_ws, size_t ws_size,
                              hipStream_t stream) {
    const float* emb    = (const float*)d_in[0];
    const int*   labels = (const int*)  d_in[1];
    // d_in[2] = num_classes scalar (compile-time 100 here)
    const float* ref_d  = (const float*)d_in[3];
    const float* ref_a  = (const float*)d_in[4];
    const float* ref_s  = (const float*)d_in[5];
    const int N = in_sizes[0] / DIMS;

    float* ws        = (float*)d_ws;
    float* psums     = ws;                                        // NB1 * 12800
    float* pcnts     = psums + (size_t)NB1 * C_CLASSES * DIMS;    // NB1 * 100
    float* centroids = pcnts + (size_t)NB1 * C_CLASSES;           // 100 * 128
    float* cn        = centroids + C_CLASSES * DIMS;              // 112 * 128
    float* counts    = cn + CPAD * DIMS;                          // 128
    float* pspread   = counts + 128;                              // NB3 * 100
    float* spreads   = pspread + (size_t)NB3 * C_CLASSES;         // 128
    float* simsP     = spreads + 128;                             // 112 * 112

    k1_seg_sums <<<NB1, DIMS, 0, stream>>>(emb, labels, N, psums, pcnts);
    k2_centroids<<<CPAD, DIMS, 0, stream>>>(psums, pcnts, centroids, cn, counts);
    k3_spread   <<<NB3, 256, 0, stream>>>(emb, labels, centroids, N, pspread);
    k5_gram_wmma<<<(CPAD / 16) * (CPAD / 16), 32, 0, stream>>>(cn, simsP);
    k4_spreads  <<<1, 128, 0, stream>>>(pspread, counts, spreads);
    k6_loss     <<<1, 256, 0, stream>>>(simsP, spreads, ref_d, ref_a, ref_s,
                                        (float*)d_out);
}